// Blocks_36945308680629
// MI455X (gfx1250) — compile-verified
//
#include <hip/hip_runtime.h>

typedef float v2f __attribute__((ext_vector_type(2)));
typedef float v8f __attribute__((ext_vector_type(8)));

#define BN_EPS 1e-5f

// D = A(16x4 f32) * B(4x16 f32) + C(16x16 f32), full f32 tensor-core math.
// A-frag lane(col,half): v0 = A[M=col][K=2*half], v1 = A[M=col][K=2*half+1]
// B-frag lane(col,half): v0 = B[K=2*half][N=col], v1 = B[K=2*half+1][N=col]
// C/D   lane(col,half) reg r: element [M = r + 8*half][N = col]
__device__ __forceinline__ v8f wmma_f32_16x16x4(v2f a, v2f b, v8f c) {
    return __builtin_amdgcn_wmma_f32_16x16x4_f32(
        /*neg_a=*/false, a, /*neg_b=*/false, b,
        /*c_mod=*/(short)0, c, /*reuse_a=*/false, /*reuse_b=*/false);
}

// ---------------------------------------------------------------------------
// zero the per-channel statistic accumulators (sum / sumsq for 6 BN stages)
__global__ void zero_stats_kernel(float* __restrict__ p, int n) {
    int i = blockIdx.x * blockDim.x + threadIdx.x;
    if (i < n) p[i] = 0.f;
}

// ---------------------------------------------------------------------------
// Depthwise 3x3 conv (pad=1) + bias. The previous stage's BN+ReLU is folded in
// as per-channel affine+ReLU on the input (PRE). Fused per-channel sum/sumsq.
template <int STRIDE, bool PRE>
__global__ void dw_kernel(const float* __restrict__ in,
                          const float* __restrict__ dww,
                          const float* __restrict__ dwb,
                          const float* __restrict__ preS,
                          const float* __restrict__ preH,
                          float* __restrict__ y1,
                          float* __restrict__ stSum,
                          float* __restrict__ stSq,
                          int C, int IH, int IW, int OH, int OW) {
    const int c = blockIdx.y % C;
    const int n = blockIdx.y / C;
    const int HWo = OH * OW;
    const int idx = blockIdx.x * blockDim.x + threadIdx.x;
    const bool valid = idx < HWo;

    float wk[9];
#pragma unroll
    for (int i = 0; i < 9; ++i) wk[i] = dww[c * 9 + i];
    float sc = 1.f, sh = 0.f;
    if (PRE) { sc = preS[c]; sh = preH[c]; }

    const float* ip = in + (size_t)(n * C + c) * (size_t)(IH * IW);

    float acc = 0.f;
    if (valid) {
        const int oh = idx / OW, ow = idx % OW;
        acc = dwb[c];
#pragma unroll
        for (int ky = 0; ky < 3; ++ky) {
            const int iy = oh * STRIDE - 1 + ky;
            if ((unsigned)iy < (unsigned)IH) {
#pragma unroll
                for (int kx = 0; kx < 3; ++kx) {
                    const int ix = ow * STRIDE - 1 + kx;
                    if ((unsigned)ix < (unsigned)IW) {
                        float v = ip[iy * IW + ix];
                        if (PRE) v = fmaxf(fmaf(sc, v, sh), 0.f);
                        acc += v * wk[ky * 3 + kx];
                    }
                }
            }
        }
        y1[(size_t)(n * C + c) * HWo + idx] = acc;
    }

    // fused per-channel statistics (channel uniform across the block)
    float s = valid ? acc : 0.f;
    float q = s * s;
#pragma unroll
    for (int off = 1; off < 32; off <<= 1) {   // wave32 reduction
        s += __shfl_xor(s, off);
        q += __shfl_xor(q, off);
    }
    __shared__ float lsum, lsq;
    if (threadIdx.x == 0) { lsum = 0.f; lsq = 0.f; }
    __syncthreads();
    if ((threadIdx.x & 31) == 0) { atomicAdd(&lsum, s); atomicAdd(&lsq, q); }
    __syncthreads();
    if (threadIdx.x == 0) { atomicAdd(&stSum[c], lsum); atomicAdd(&stSq[c], lsq); }
}

// ---------------------------------------------------------------------------
// Pointwise 1x1 conv as WMMA GEMM.  M = cout (16 per wave), N = 64 positions
// (4 N-tiles reusing the weight A-fragment), K = cin in steps of 4.
//
// The activation panel (CIN x 64 positions) is staged through LDS once per
// panel-group: BN1+ReLU applied during the cooperative b128 load, then the
// inner loop is ds_load + one L2-hot b64 weight load + WMMA.
// A block runs 8 waves = PANELS (= 8/CT) panels x CT cout-tiles.
template <int CIN, int COUT>
__launch_bounds__(256)
__global__ void pw_kernel(const float* __restrict__ y1,
                          const float* __restrict__ w,      // (COUT, CIN)
                          const float* __restrict__ bias,
                          const float* __restrict__ s1,
                          const float* __restrict__ h1,
                          float* __restrict__ y2,
                          float* __restrict__ stSum,
                          float* __restrict__ stSq,
                          int HW, int N) {
    constexpr int CT     = COUT / 16;
    constexpr int PANELS = 8 / CT;            // panels per block (8 waves)
    constexpr int PANSZ  = CIN * 64;          // floats per panel

    extern __shared__ float smem[];           // PANELS * PANSZ floats
    __shared__ float lsum[COUT];
    __shared__ float lsq[COUT];
    for (int t = threadIdx.x; t < COUT; t += blockDim.x) { lsum[t] = 0.f; lsq[t] = 0.f; }

    const int PB = HW / 64;                   // 64-position blocks per image

    // ---- cooperative panel load: global (b128, coalesced) -> BN+ReLU -> LDS
    constexpr int VCHUNKS = (PANELS * PANSZ) / 4;   // float4 chunks
#pragma unroll
    for (int it = 0; it < VCHUNKS / 256; ++it) {
        const int cid = it * 256 + threadIdx.x;
        const int e   = cid * 4;
        const int pan = e / PANSZ;
        const int rem = e % PANSZ;
        const int ci  = rem / 64;
        const int pos = rem % 64;
        const int mbp = blockIdx.x * PANELS + pan;
        const int n   = mbp / PB;
        const int p0  = (mbp % PB) * 64;
        const float sc = s1[ci], sh = h1[ci];
        const float4 v =
            *(const float4*)(y1 + (size_t)(n * CIN + ci) * HW + p0 + pos);
        float4 r;
        r.x = fmaxf(fmaf(sc, v.x, sh), 0.f);
        r.y = fmaxf(fmaf(sc, v.y, sh), 0.f);
        r.z = fmaxf(fmaf(sc, v.z, sh), 0.f);
        r.w = fmaxf(fmaf(sc, v.w, sh), 0.f);
        *(float4*)(smem + pan * PANSZ + ci * 64 + pos) = r;
    }
    __syncthreads();

    // ---- compute phase -----------------------------------------------------
    const int wave = threadIdx.x >> 5;        // 8 waves / block (wave32)
    const int lane = threadIdx.x & 31;
    const int col  = lane & 15;
    const int half = lane >> 4;
    const int pan  = wave / CT;
    const int ct   = wave % CT;
    const int mb   = blockIdx.x * PANELS + pan;

    if (mb < N * PB) {                        // wave-uniform: EXEC all-ones
        const int n  = mb / PB;
        const int p0 = (mb % PB) * 64;
        const float* arow = w + (size_t)(ct * 16 + col) * CIN;  // A: M=cout
        const float* pbase = smem + pan * PANSZ;

        v8f acc[4];
#pragma unroll
        for (int nt = 0; nt < 4; ++nt)
#pragma unroll
            for (int r = 0; r < 8; ++r) acc[nt][r] = 0.f;

        for (int k0 = 0; k0 < CIN; k0 += 4) {
            const int ka = k0 + 2 * half;
            v2f a;
            a.x = arow[ka];
            a.y = arow[ka + 1];
            const float* l0 = pbase + ka * 64 + col;
#pragma unroll
            for (int nt = 0; nt < 4; ++nt) {
                v2f b;
                b.x = l0[nt * 16];
                b.y = l0[64 + nt * 16];
                acc[nt] = wmma_f32_16x16x4(a, b, acc[nt]);
            }
        }

        // epilogue: bias, coalesced store (N=pos striped across lanes), stats
#pragma unroll
        for (int r = 0; r < 8; ++r) {
            const int co = ct * 16 + r + 8 * half;
            const float bv = bias[co];
            float* orow = y2 + (size_t)(n * COUT + co) * HW + p0 + col;
            float s = 0.f, q = 0.f;
#pragma unroll
            for (int nt = 0; nt < 4; ++nt) {
                const float v = acc[nt][r] + bv;
                orow[nt * 16] = v;
                s += v;
                q += v * v;
            }
#pragma unroll
            for (int off = 1; off < 16; off <<= 1) {  // reduce across 16 cols
                s += __shfl_xor(s, off);
                q += __shfl_xor(q, off);
            }
            if (col == 0) { atomicAdd(&lsum[co], s); atomicAdd(&lsq[co], q); }
        }
    }
    __syncthreads();
    for (int t = threadIdx.x; t < COUT; t += blockDim.x) {
        atomicAdd(&stSum[t], lsum[t]);
        atomicAdd(&stSq[t], lsq[t]);
    }
}

// ---------------------------------------------------------------------------
// (sum, sumsq, gamma, beta) -> per-channel (scale, shift) for lazy BN
__global__ void finalize_kernel(const float* __restrict__ sum,
                                const float* __restrict__ sq,
                                const float* __restrict__ g,
                                const float* __restrict__ b,
                                float* __restrict__ scale,
                                float* __restrict__ shift,
                                int C, float invCnt) {
    const int t = blockIdx.x * blockDim.x + threadIdx.x;
    if (t < C) {
        const float m   = sum[t] * invCnt;
        const float var = sq[t] * invCnt - m * m;
        const float s   = g[t] * rsqrtf(var + BN_EPS);
        scale[t] = s;
        shift[t] = b[t] - s * m;
    }
}

// ---------------------------------------------------------------------------
// final BN+ReLU + global average pool; one block per (n, c) plane
__global__ void avg_kernel(const float* __restrict__ y,
                           const float* __restrict__ s2,
                           const float* __restrict__ h2,
                           float* __restrict__ out,
                           int C, int HW) {
    const int c = blockIdx.x % C;
    const float sc = s2[c], sh = h2[c];
    const float* p = y + (size_t)blockIdx.x * HW;
    float s = 0.f;
    for (int i = threadIdx.x; i < HW; i += blockDim.x)
        s += fmaxf(fmaf(sc, p[i], sh), 0.f);
#pragma unroll
    for (int off = 1; off < 32; off <<= 1) s += __shfl_xor(s, off);
    __shared__ float wsum[8];
    if ((threadIdx.x & 31) == 0) wsum[threadIdx.x >> 5] = s;
    __syncthreads();
    if (threadIdx.x == 0) {
        float t = 0.f;
#pragma unroll
        for (int i = 0; i < 8; ++i) t += wsum[i];
        out[blockIdx.x] = t / (float)HW;
    }
}

// ---------------------------------------------------------------------------
extern "C" void kernel_launch(void* const* d_in, const int* in_sizes, int n_in,
                              void* d_out, int out_size, void* d_ws, size_t ws_size,
                              hipStream_t stream) {
    (void)in_sizes; (void)n_in; (void)out_size; (void)ws_size;
    const float* x = (const float*)d_in[0];
    const float *dw_w[3], *dw_b[3], *g1[3], *be1[3], *pw_w[3], *pw_b[3], *g2[3], *be2[3];
    for (int b = 0; b < 3; ++b) {
        const int base = 1 + b * 8;
        dw_w[b] = (const float*)d_in[base + 0];
        dw_b[b] = (const float*)d_in[base + 1];
        g1[b]   = (const float*)d_in[base + 2];
        be1[b]  = (const float*)d_in[base + 3];
        pw_w[b] = (const float*)d_in[base + 4];
        pw_b[b] = (const float*)d_in[base + 5];
        g2[b]   = (const float*)d_in[base + 6];
        be2[b]  = (const float*)d_in[base + 7];
    }

    // workspace layout (byte offsets, 256B aligned)
    char* ws = (char*)d_ws;
    float* stats  = (float*)ws;                 // 6 stages * 256 floats (sum|sq)
    float* ssbase = (float*)(ws + 8192);        // 6 stages * 256 floats (scale|shift)
    const size_t SZ_C = 6422528;                // 32*64*56*56   elems (25.7 MB)
    const size_t SZ_A = 12845056;               // 32*32*112*112 elems (51.4 MB)
    float* bufC = (float*)(ws + 16384);                        // y1_1
    float* bufA = bufC + SZ_C;                                 // y1_0, later y2_1
    float* bufB = bufA + SZ_A;                                 // y2_0, later y1_2|y2_2
    float* y1_2 = bufB;
    float* y2_2 = bufB + SZ_A;

    auto SUM = [&](int s) { return stats  + s * 256; };
    auto SQ  = [&](int s) { return stats  + s * 256 + 128; };
    auto SC  = [&](int s) { return ssbase + s * 256; };
    auto SH  = [&](int s) { return ssbase + s * 256 + 128; };

    const int N = 32;
    const float inv0 = 1.f / 401408.f;  // N*112*112
    const float inv1 = 1.f / 100352.f;  // N*56*56

    zero_stats_kernel<<<6, 256, 0, stream>>>(stats, 1536);

    // ---- block 0: 32 -> 64, stride 1, 112x112 ----
    dw_kernel<1, false><<<dim3(49, N * 32), 256, 0, stream>>>(
        x, dw_w[0], dw_b[0], nullptr, nullptr, bufA, SUM(0), SQ(0),
        32, 112, 112, 112, 112);
    finalize_kernel<<<1, 128, 0, stream>>>(SUM(0), SQ(0), g1[0], be1[0], SC(0), SH(0), 32, inv0);
    pw_kernel<32, 64><<<3136, 256, 2 * 32 * 64 * 4, stream>>>(
        bufA, pw_w[0], pw_b[0], SC(0), SH(0), bufB, SUM(1), SQ(1), 12544, N);
    finalize_kernel<<<1, 128, 0, stream>>>(SUM(1), SQ(1), g2[0], be2[0], SC(1), SH(1), 64, inv0);

    // ---- block 1: 64 -> 128, stride 2, 112x112 -> 56x56 ----
    dw_kernel<2, true><<<dim3(13, N * 64), 256, 0, stream>>>(
        bufB, dw_w[1], dw_b[1], SC(1), SH(1), bufC, SUM(2), SQ(2),
        64, 112, 112, 56, 56);
    finalize_kernel<<<1, 128, 0, stream>>>(SUM(2), SQ(2), g1[1], be1[1], SC(2), SH(2), 64, inv1);
    pw_kernel<64, 128><<<1568, 256, 1 * 64 * 64 * 4, stream>>>(
        bufC, pw_w[1], pw_b[1], SC(2), SH(2), bufA, SUM(3), SQ(3), 3136, N);
    finalize_kernel<<<1, 128, 0, stream>>>(SUM(3), SQ(3), g2[1], be2[1], SC(3), SH(3), 128, inv1);

    // ---- block 2: 128 -> 128, stride 1, 56x56 ----
    dw_kernel<1, true><<<dim3(13, N * 128), 256, 0, stream>>>(
        bufA, dw_w[2], dw_b[2], SC(3), SH(3), y1_2, SUM(4), SQ(4),
        128, 56, 56, 56, 56);
    finalize_kernel<<<1, 128, 0, stream>>>(SUM(4), SQ(4), g1[2], be1[2], SC(4), SH(4), 128, inv1);
    pw_kernel<128, 128><<<1568, 256, 1 * 128 * 64 * 4, stream>>>(
        y1_2, pw_w[2], pw_b[2], SC(4), SH(4), y2_2, SUM(5), SQ(5), 3136, N);
    finalize_kernel<<<1, 128, 0, stream>>>(SUM(5), SQ(5), g2[2], be2[2], SC(5), SH(5), 128, inv1);

    // ---- final BN+ReLU + global average pool -> (32,128,1,1) ----
    avg_kernel<<<N * 128, 256, 0, stream>>>(y2_2, SC(5), SH(5), (float*)d_out, 128, 3136);
}